// LASI_21517786153235
// MI455X (gfx1250) — compile-verified
//
#include <hip/hip_runtime.h>

typedef __attribute__((ext_vector_type(2))) float v2f;
typedef __attribute__((ext_vector_type(8))) float v8f;

#define Nn    4096
#define Kn    12
#define GRAM  78            // upper triangle of 12x12
#define BCOLS 96            // 78 gram + 12 target + 6 pad
#define NGRP  (Nn / 4)      // K-groups of 4 rows
#define GRPF  (6 * 64)      // floats per K-group in swizzled B: 6 frags * 32 lanes * 2
#define REGV  (80.0f / 127.5f)            // REG=1.0 * REG_SCALE
#define LOG2_DECAY (-0.3219280948873623f) // log2(0.8)

// ---------------------------------------------------------------------------
// Phase 1: build RHS matrix B directly in WMMA-fragment-swizzled order.
// Logical B (N x 96) row i: [outer-upper-tri(78) | target(12) | 0(6)].
// Swizzled layout: group g=j/4, frag cb=col/16, lane = kh*16 + (col&15) with
// kh=(j&3)>>1, elem=(j&1):  addr = g*384 + cb*64 + lane*2 + elem.
// A wave then fetches one fragment with a single coalesced b64 load.
// ---------------------------------------------------------------------------
__global__ void build_B(const float* __restrict__ x, const float* __restrict__ y,
                        const int* __restrict__ mask,
                        float* __restrict__ Bx, float* __restrict__ By) {
  int i = blockIdx.x * blockDim.x + threadIdx.x;   // logical B row j == i
  if (i >= Nn) return;
  const float* tf = blockIdx.y ? y : x;
  float* B = blockIdx.y ? By : Bx;

  float nb[Kn];
#pragma unroll
  for (int k = 0; k < Kn; ++k) {
    int idx = mask[i * Kn + k];
    nb[k] = (idx >= 0) ? tf[idx] : 0.0f;
  }

  const int g  = i >> 2;
  const int kh = (i & 3) >> 1;
  const int e  = i & 1;
  float* base = B + (size_t)g * GRPF + (size_t)kh * 32 + e;
  // column c lands at base[(c>>4)*64 + (c&15)*2]
  int c = 0;
#pragma unroll
  for (int k = 0; k < Kn; ++k) {
#pragma unroll
    for (int l = k; l < Kn; ++l) {
      base[(c >> 4) * 64 + (c & 15) * 2] = nb[k] * nb[l];
      ++c;
    }
  }
  float ti = tf[i];
#pragma unroll
  for (int k = 0; k < Kn; ++k) {
    int cc = GRAM + k;
    base[(cc >> 4) * 64 + (cc & 15) * 2] = ti * nb[k];
  }
#pragma unroll
  for (int cc = 90; cc < BCOLS; ++cc)
    base[(cc >> 4) * 64 + (cc & 15) * 2] = 0.0f;
}

// ---------------------------------------------------------------------------
// Phase 2: D = W @ B  via V_WMMA_F32_16X16X4_F32, W synthesized in registers.
// One wave per 16-row tile, 6 column blocks (96 cols) per wave.
// A frag (16x4 f32): lanes 0-15 hold K=0,1 ; lanes 16-31 hold K=2,3.
// B frags come pre-swizzled: one global_load_b64 per fragment.
// C/D (16x16 f32): VGPR r, lanes 0-15 -> M=r, lanes 16-31 -> M=r+8, N=lane&15.
// ---------------------------------------------------------------------------
__global__ void __launch_bounds__(32)
wgemm(const float* __restrict__ Bx, const float* __restrict__ By,
      float* __restrict__ Dx, float* __restrict__ Dy) {
  const float* B = blockIdx.y ? By : Bx;
  float* D = blockIdx.y ? Dy : Dx;

  const int lane = threadIdx.x;       // 0..31, EXEC all ones (no divergence)
  const int m  = lane & 15;
  const int kh = lane >> 4;           // 0: K pair {0,1}; 1: K pair {2,3}
  const int i  = blockIdx.x * 16 + m; // global row this lane's A-values belong to
  const int ri = i >> 6, ci = i & 63;

  v8f acc0 = {}, acc1 = {}, acc2 = {}, acc3 = {}, acc4 = {}, acc5 = {};

  for (int k = 0; k < Nn; k += 4) {
    const int j0 = k + (kh << 1);

    // --- A fragment: W[i, j0], W[i, j0+1] = causal * 0.8^l1 ----------------
    v2f a;
    {
      int j = j0;
      int rj = j >> 6, cj = j & 63;
      int dr = ri - rj; dr = dr < 0 ? -dr : dr;
      int dc = ci - cj; dc = dc < 0 ? -dc : dc;
      float w = __builtin_amdgcn_exp2f((float)(dr + dc) * LOG2_DECAY);
      a.x = (j < i) ? w : 0.0f;

      j = j0 + 1;
      rj = j >> 6; cj = j & 63;
      dr = ri - rj; dr = dr < 0 ? -dr : dr;
      dc = ci - cj; dc = dc < 0 ? -dc : dc;
      w = __builtin_amdgcn_exp2f((float)(dr + dc) * LOG2_DECAY);
      a.y = (j < i) ? w : 0.0f;
    }

    // --- B fragments: pre-swizzled, one b64 per fragment -------------------
    const float* Bg = B + (size_t)(k >> 2) * GRPF + (size_t)lane * 2;
    __builtin_prefetch(Bg + 8 * GRPF, 0, 1);  // speculative stream prefetch

#define WMMA_CB(cb, accv)                                                     \
    {                                                                         \
      v2f b = *(const v2f*)(Bg + (cb) * 64);                                  \
      accv = __builtin_amdgcn_wmma_f32_16x16x4_f32(                           \
          false, a, false, b, (short)0, accv, false, false);                  \
    }
    WMMA_CB(0, acc0) WMMA_CB(1, acc1) WMMA_CB(2, acc2)
    WMMA_CB(3, acc3) WMMA_CB(4, acc4) WMMA_CB(5, acc5)
#undef WMMA_CB
  }

  // --- store D tile (row-major D, N x 96) ---------------------------------
  float* Dr = D + (size_t)(blockIdx.x * 16 + kh * 8) * BCOLS + m;
#pragma unroll
  for (int r = 0; r < 8; ++r) {
    float* p = Dr + (size_t)r * BCOLS;
    p[0]  = acc0[r];
    p[16] = acc1[r];
    p[32] = acc2[r];
    p[48] = acc3[r];
    p[64] = acc4[r];
    p[80] = acc5[r];
  }
}

// ---------------------------------------------------------------------------
// Phase 3: per-element 12x12 SPD solve (Cholesky) + unit normalization.
// coef = unpack(D[:,0:78]) + REGV*I ; tgt = D[:,78:90]
// ---------------------------------------------------------------------------
__global__ void solve(const float* __restrict__ Dx, const float* __restrict__ Dy,
                      float* __restrict__ px, float* __restrict__ py) {
  int i = blockIdx.x * blockDim.x + threadIdx.x;
  if (i >= Nn) return;
  const float* row = (blockIdx.y ? Dy : Dx) + (size_t)i * BCOLS;
  float* P = blockIdx.y ? py : px;

  float A[Kn][Kn];
  float b[Kn];
  {
    int c = 0;
#pragma unroll
    for (int k = 0; k < Kn; ++k) {
#pragma unroll
      for (int l = k; l < Kn; ++l) {
        float v = row[c++];
        A[k][l] = v;
        A[l][k] = v;
      }
      A[k][k] += REGV;
    }
#pragma unroll
    for (int k = 0; k < Kn; ++k) b[k] = row[GRAM + k];
  }

  // Cholesky: A = L L^T (stored in lower triangle of A)
#pragma unroll
  for (int k = 0; k < Kn; ++k) {
    float d = A[k][k];
#pragma unroll
    for (int p = 0; p < k; ++p) d -= A[k][p] * A[k][p];
    d = sqrtf(fmaxf(d, 1e-20f));
    A[k][k] = d;
    float inv = 1.0f / d;
#pragma unroll
    for (int r = k + 1; r < Kn; ++r) {
      float s = A[r][k];
#pragma unroll
      for (int p = 0; p < k; ++p) s -= A[r][p] * A[k][p];
      A[r][k] = s * inv;
    }
  }
  // forward substitution: L z = b
#pragma unroll
  for (int k = 0; k < Kn; ++k) {
    float s = b[k];
#pragma unroll
    for (int p = 0; p < k; ++p) s -= A[k][p] * b[p];
    b[k] = s / A[k][k];
  }
  // backward substitution: L^T w = z
#pragma unroll
  for (int k = Kn - 1; k >= 0; --k) {
    float s = b[k];
#pragma unroll
    for (int p = k + 1; p < Kn; ++p) s -= A[p][k] * b[p];
    b[k] = s / A[k][k];
  }
  // unit normalize
  float nrm = 0.0f;
#pragma unroll
  for (int k = 0; k < Kn; ++k) nrm += b[k] * b[k];
  float invn = 1.0f / (sqrtf(nrm) + 1e-8f);
#pragma unroll
  for (int k = 0; k < Kn; ++k) P[(size_t)i * Kn + k] = b[k] * invn;
}

// ---------------------------------------------------------------------------
// Phase 4: out = || px - py ||  (single-block reduction)
// ---------------------------------------------------------------------------
__global__ void reduce_dist(const float* __restrict__ px,
                            const float* __restrict__ py,
                            float* __restrict__ out) {
  __shared__ float sm[256];
  float s = 0.0f;
  for (int idx = threadIdx.x; idx < Nn * Kn; idx += 256) {
    float d = px[idx] - py[idx];
    s += d * d;
  }
  sm[threadIdx.x] = s;
  __syncthreads();
  for (int off = 128; off > 0; off >>= 1) {
    if (threadIdx.x < off) sm[threadIdx.x] += sm[threadIdx.x + off];
    __syncthreads();
  }
  if (threadIdx.x == 0) out[0] = sqrtf(sm[0]);
}

// ---------------------------------------------------------------------------
extern "C" void kernel_launch(void* const* d_in, const int* in_sizes, int n_in,
                              void* d_out, int out_size, void* d_ws, size_t ws_size,
                              hipStream_t stream) {
  const float* x    = (const float*)d_in[0];
  const float* y    = (const float*)d_in[1];
  const int*   mask = (const int*)d_in[2];

  float* ws = (float*)d_ws;
  float* Bx = ws;                      // swizzled, 4096*96 floats
  float* By = Bx + (size_t)Nn * BCOLS;
  float* Dx = By + (size_t)Nn * BCOLS; // row-major, 4096*96 floats
  float* Dy = Dx + (size_t)Nn * BCOLS;
  float* px = Dy + (size_t)Nn * BCOLS;
  float* py = px + (size_t)Nn * Kn;    // total ~6.7 MB

  build_B    <<<dim3(Nn / 256, 2), 256, 0, stream>>>(x, y, mask, Bx, By);
  wgemm      <<<dim3(Nn / 16, 2),  32,  0, stream>>>(Bx, By, Dx, Dy);
  solve      <<<dim3(Nn / 256, 2), 256, 0, stream>>>(Dx, Dy, px, py);
  reduce_dist<<<1, 256, 0, stream>>>(px, py, (float*)d_out);
}